// EmbResGCNBlock_3582002725001
// MI455X (gfx1250) — compile-verified
//
#include <hip/hip_runtime.h>
#include <hip/hip_bf16.h>

#define DFEAT 128
#define BN_EPS 1e-5f

typedef __attribute__((ext_vector_type(2))) float v2f;
typedef __attribute__((ext_vector_type(8))) float v8f;

// ---------------------------------------------------------------------------
// K1: agg = (1 + eps) * x   (GIN self term), float4 vectorized
// ---------------------------------------------------------------------------
__global__ void k_init_agg(const float* __restrict__ x, const float* __restrict__ eps_p,
                           float* __restrict__ agg, int n4) {
  int i = blockIdx.x * blockDim.x + threadIdx.x;
  if (i >= n4) return;
  const float s = 1.0f + eps_p[0];
  float4 v = reinterpret_cast<const float4*>(x)[i];
  v.x *= s; v.y *= s; v.z *= s; v.w *= s;
  reinterpret_cast<float4*>(agg)[i] = v;
}

// ---------------------------------------------------------------------------
// K2: zero 512-float stats slab (sum | sumsq | scale | shift)
// ---------------------------------------------------------------------------
__global__ void k_zero_stats(float* __restrict__ stats) {
  stats[threadIdx.x] = 0.0f;
}

// ---------------------------------------------------------------------------
// K3: edge scatter-add. One wave32 per edge; lane owns 4 consecutive features.
// agg[dst][:] += x[src][:]  via global_atomic_add_f32 (resolves in L2).
// ---------------------------------------------------------------------------
__global__ void k_edge_scatter(const float* __restrict__ x, const long long* __restrict__ ei,
                               float* __restrict__ agg, int n_edges) {
  int t = blockIdx.x * blockDim.x + threadIdx.x;
  int e = t >> 5;
  if (e >= n_edges) return;
  int lane = t & 31;
  const long long s = ei[e];            // src row
  const long long d = ei[n_edges + e];  // dst row
  float4 v = reinterpret_cast<const float4*>(x + (size_t)s * DFEAT)[lane];
  float* dst = agg + (size_t)d * DFEAT + lane * 4;
  atomicAdd(dst + 0, v.x);
  atomicAdd(dst + 1, v.y);
  atomicAdd(dst + 2, v.z);
  atomicAdd(dst + 3, v.w);
}

// ---------------------------------------------------------------------------
// K4: h = agg @ W^T + b using V_WMMA_F32_16X16X4_F32 (fp32 matrix pipe).
// One wave per 16x16 output tile; 8 waves/block cover all 128 columns of a
// 16-row stripe. A-layout (ISA 7.12.2, 32-bit 16x4): lane L holds
// A[M=L&15][K = k + 2*(L>>4) + j] in vgpr j. B (4x16) mirrors it; since
// B = W^T tile and W is [f][d] row-major, both sides load a float2 at
// row (L&15), offset k + 2*(L>>4).
// C/D layout: acc[v] -> row = v + 8*(lane>>4), col = lane&15.
// ---------------------------------------------------------------------------
__global__ void k_gemm_wmma(const float* __restrict__ agg, const float* __restrict__ W,
                            const float* __restrict__ bias, float* __restrict__ h) {
  const int lane = threadIdx.x & 31;
  const int wave = threadIdx.x >> 5;        // 0..7 -> column tile
  const int row0 = blockIdx.x * 16;
  const int col0 = wave * 16;
  const int r    = lane & 15;
  const int koff = (lane >> 4) * 2;

  const float* aptr = agg + (size_t)(row0 + r) * DFEAT + koff;
  const float* bptr = W   + (size_t)(col0 + r) * DFEAT + koff;

  v8f acc = {};
#pragma unroll
  for (int k = 0; k < DFEAT; k += 4) {
    v2f a = *reinterpret_cast<const v2f*>(aptr + k);
    v2f b = *reinterpret_cast<const v2f*>(bptr + k);
    acc = __builtin_amdgcn_wmma_f32_16x16x4_f32(
        /*neg_a=*/false, a, /*neg_b=*/false, b,
        /*c_mod=*/(short)0, acc, /*reuse_a=*/false, /*reuse_b=*/false);
  }

  const float bv  = bias[col0 + r];
  const int mbase = row0 + ((lane >> 4) << 3);
  float* out = h + (size_t)mbase * DFEAT + col0 + r;
#pragma unroll
  for (int v = 0; v < 8; ++v) out[(size_t)v * DFEAT] = acc[v] + bv;
}

// ---------------------------------------------------------------------------
// K5: per-feature column sums of h (coalesced full-row reads per block)
// ---------------------------------------------------------------------------
__global__ void k_colstats(const float* __restrict__ h, float* __restrict__ sum,
                           float* __restrict__ sumsq, int n_nodes) {
  const int f = threadIdx.x;  // 128 threads = one row
  float s = 0.0f, ss = 0.0f;
  for (int r = blockIdx.x; r < n_nodes; r += gridDim.x) {
    float v = h[(size_t)r * DFEAT + f];
    s += v;
    ss += v * v;
  }
  atomicAdd(&sum[f], s);
  atomicAdd(&sumsq[f], ss);
}

// ---------------------------------------------------------------------------
// K6: fold BN stats into affine scale/shift per feature
// ---------------------------------------------------------------------------
__global__ void k_finalize_stats(const float* __restrict__ sum, const float* __restrict__ sumsq,
                                 const float* __restrict__ gamma, const float* __restrict__ beta,
                                 float* __restrict__ scale, float* __restrict__ shift,
                                 int n_nodes) {
  const int f = threadIdx.x;  // 128
  const float invN = 1.0f / (float)n_nodes;
  const float mean = sum[f] * invN;
  const float var  = sumsq[f] * invN - mean * mean;
  const float sc   = gamma[f] * rsqrtf(var + BN_EPS);
  scale[f] = sc;
  shift[f] = beta[f] - mean * sc;
}

// ---------------------------------------------------------------------------
// K7: out = relu(h*scale + shift) + x, float4 vectorized, in-place on d_out
// ---------------------------------------------------------------------------
__global__ void k_bn_relu_res(const float* __restrict__ x, const float* __restrict__ scale,
                              const float* __restrict__ shift, float* __restrict__ out, int n4) {
  int i = blockIdx.x * blockDim.x + threadIdx.x;
  if (i >= n4) return;
  const int c = i & (DFEAT / 4 - 1);  // feature-quad index 0..31
  float4 hv = reinterpret_cast<float4*>(out)[i];
  float4 xv = reinterpret_cast<const float4*>(x)[i];
  float4 sc = reinterpret_cast<const float4*>(scale)[c];
  float4 sh = reinterpret_cast<const float4*>(shift)[c];
  float4 r;
  r.x = fmaxf(hv.x * sc.x + sh.x, 0.0f) + xv.x;
  r.y = fmaxf(hv.y * sc.y + sh.y, 0.0f) + xv.y;
  r.z = fmaxf(hv.z * sc.z + sh.z, 0.0f) + xv.z;
  r.w = fmaxf(hv.w * sc.w + sh.w, 0.0f) + xv.w;
  reinterpret_cast<float4*>(out)[i] = r;
}

// ---------------------------------------------------------------------------
extern "C" void kernel_launch(void* const* d_in, const int* in_sizes, int n_in,
                              void* d_out, int out_size, void* d_ws, size_t ws_size,
                              hipStream_t stream) {
  const float*     x     = (const float*)d_in[0];
  const long long* ei    = (const long long*)d_in[1];  // int64 edge_index [2, E]
  const float*     W     = (const float*)d_in[2];
  const float*     b     = (const float*)d_in[3];
  const float*     eps   = (const float*)d_in[4];
  const float*     gamma = (const float*)d_in[5];
  const float*     beta  = (const float*)d_in[6];
  float*           out   = (float*)d_out;

  const int n_nodes = in_sizes[0] / DFEAT;
  const int n_edges = in_sizes[1] / 2;
  const int n4      = n_nodes * (DFEAT / 4);

  // workspace layout: agg [n_nodes*DFEAT] floats, then 512-float stats slab
  float* agg   = (float*)d_ws;
  float* stats = agg + (size_t)n_nodes * DFEAT;
  float* sum   = stats;
  float* sumsq = stats + DFEAT;
  float* scale = stats + 2 * DFEAT;
  float* shift = stats + 3 * DFEAT;

  k_init_agg<<<(n4 + 255) / 256, 256, 0, stream>>>(x, eps, agg, n4);
  k_zero_stats<<<1, 4 * DFEAT, 0, stream>>>(stats);
  k_edge_scatter<<<(n_edges * 32 + 255) / 256, 256, 0, stream>>>(x, ei, agg, n_edges);
  k_gemm_wmma<<<n_nodes / 16, 256, 0, stream>>>(agg, W, b, out);
  k_colstats<<<1024, DFEAT, 0, stream>>>(out, sum, sumsq, n_nodes);
  k_finalize_stats<<<1, DFEAT, 0, stream>>>(sum, sumsq, gamma, beta, scale, shift, n_nodes);
  k_bn_relu_res<<<(n4 + 255) / 256, 256, 0, stream>>>(x, scale, shift, out, n4);
}